// GCNModel_42657615184421
// MI455X (gfx1250) — compile-verified
//
#include <hip/hip_runtime.h>

typedef __attribute__((ext_vector_type(2))) float v2f;
typedef __attribute__((ext_vector_type(8))) float v8f;

// ---------------------------------------------------------------------------
// Degree / normalization: deg[i] = 1 (self loop) + #incoming edges; dinv = rsqrt
// ---------------------------------------------------------------------------
__global__ void k_deg_init(float* __restrict__ deg, int n) {
  int i = blockIdx.x * blockDim.x + threadIdx.x;
  if (i < n) deg[i] = 1.0f;  // self-loop contribution
}

__global__ void k_deg_count(const int* __restrict__ dst, float* __restrict__ deg, int E) {
  int e = blockIdx.x * blockDim.x + threadIdx.x;
  if (e < E) unsafeAtomicAdd(&deg[dst[e]], 1.0f);
}

__global__ void k_dinv(float* __restrict__ deg, int n) {
  int i = blockIdx.x * blockDim.x + threadIdx.x;
  if (i < n) deg[i] = rsqrtf(deg[i]);   // deg >= 1 always (self loops)
}

// ---------------------------------------------------------------------------
// FP32 WMMA GEMM, fused epilogue:
//   C[M,N]   = A[M,K] * B[K,N]                (unscaled, for edge gather)
//   Acc[M,N] = C * dinv[row]^2                (self-loop accumulator init)
// K, N compile-time => all k-loop offsets are immediates (no address ALU).
// One wave computes TWO adjacent 16x16 tiles (32 cols) sharing one A fragment.
// Requires M%16==0 (100000 ok), N%32==0, EXEC all-ones per wave.
// A 16x4 frag: lanes 0-15 hold K=k,k+1 ; lanes 16-31 hold K=k+2,k+3 (per row)
// B 4x16 frag: same K split, N = lane&15
// C tile:      vgpr v -> row m0+v (lanes 0-15) / m0+8+v (lanes 16-31)
// ---------------------------------------------------------------------------
template <int K, int N>
__global__ __launch_bounds__(128) void k_gemm_wmma(
    const float* __restrict__ A, const float* __restrict__ B,
    const float* __restrict__ dinv,
    float* __restrict__ C, float* __restrict__ Acc) {
  const int lane = threadIdx.x & 31;
  const int wave = threadIdx.x >> 5;
  const int m0 = blockIdx.x << 4;
  const int n0 = wave << 5;          // 32 columns per wave; block covers N
  if (n0 >= N) return;               // uniform per wave

  const int half = lane >> 4;        // 0: lanes 0-15, 1: lanes 16-31
  const int l15  = lane & 15;

  const float* arow  = A + (size_t)(m0 + l15) * K + 2 * half;
  const float* bcol0 = B + (size_t)(2 * half) * N + (n0 + l15);
  const float* bcol1 = bcol0 + 16;

  v8f acc0 = {}, acc1 = {};
#pragma unroll
  for (int k = 0; k < K; k += 4) {
    v2f a, b0, b1;
    a.x  = arow[k];
    a.y  = arow[k + 1];
    b0.x = bcol0[k * N];
    b0.y = bcol0[(k + 1) * N];
    b1.x = bcol1[k * N];
    b1.y = bcol1[(k + 1) * N];
    acc0 = __builtin_amdgcn_wmma_f32_16x16x4_f32(false, a, false, b0,
                                                 (short)0, acc0, false, false);
    acc1 = __builtin_amdgcn_wmma_f32_16x16x4_f32(false, a, false, b1,
                                                 (short)0, acc1, false, false);
  }

  const int crow = m0 + half * 8;
  float* cp = C   + (size_t)crow * N + (n0 + l15);
  float* ap = Acc + (size_t)crow * N + (n0 + l15);
#pragma unroll
  for (int v = 0; v < 8; ++v) {
    float di = dinv[crow + v];       // uniform within each 16-lane group
    float sc = di * di;
    cp[v * N]      = acc0[v];
    cp[v * N + 16] = acc1[v];
    ap[v * N]      = acc0[v] * sc;
    ap[v * N + 16] = acc1[v] * sc;
  }
}

// ---------------------------------------------------------------------------
// Edge scatter-add: one wave per edge, vector loads, float atomics.
// acc[dst,:] += h[src,:] * dinv[src]*dinv[dst]
// ---------------------------------------------------------------------------
template <int F>
__global__ void k_edge_agg(const int* __restrict__ src, const int* __restrict__ dst,
                           const float* __restrict__ dinv, const float* __restrict__ h,
                           float* __restrict__ acc, int E) {
  int wid = blockIdx.x * (blockDim.x >> 5) + (threadIdx.x >> 5);
  int lane = threadIdx.x & 31;
  if (wid >= E) return;  // uniform per wave
  int s = src[wid];
  int d = dst[wid];
  float norm = dinv[s] * dinv[d];
  const float* hs = h + (size_t)s * F;
  float* ad = acc + (size_t)d * F;
  if constexpr (F == 128) {
    float4 v = ((const float4*)hs)[lane];          // global_load_b128
    float* p = ad + lane * 4;
    unsafeAtomicAdd(p + 0, v.x * norm);
    unsafeAtomicAdd(p + 1, v.y * norm);
    unsafeAtomicAdd(p + 2, v.z * norm);
    unsafeAtomicAdd(p + 3, v.w * norm);
  } else {  // F == 64
    float2 v = ((const float2*)hs)[lane];          // global_load_b64
    float* p = ad + lane * 2;
    unsafeAtomicAdd(p + 0, v.x * norm);
    unsafeAtomicAdd(p + 1, v.y * norm);
  }
}

// ---------------------------------------------------------------------------
// a[i,f] = relu(a[i,f] + b[f])
// ---------------------------------------------------------------------------
__global__ void k_bias_relu(float* __restrict__ a, const float* __restrict__ b,
                            long long total, int fmask) {
  long long t = (long long)blockIdx.x * blockDim.x + threadIdx.x;
  if (t < total) a[t] = fmaxf(a[t] + b[(int)(t & fmask)], 0.0f);
}

// ---------------------------------------------------------------------------
// out[i,:] = log_softmax(out[i,:] + bias) over 64 classes; one wave per node.
// ---------------------------------------------------------------------------
__global__ void k_bias_logsoftmax64(float* __restrict__ out, const float* __restrict__ bias,
                                    int n) {
  int node = blockIdx.x * (blockDim.x >> 5) + (threadIdx.x >> 5);
  int lane = threadIdx.x & 31;
  if (node >= n) return;  // uniform per wave
  float* row = out + (size_t)node * 64;
  float v0 = row[lane] + bias[lane];
  float v1 = row[lane + 32] + bias[lane + 32];
  float m = fmaxf(v0, v1);
#pragma unroll
  for (int o = 16; o; o >>= 1) m = fmaxf(m, __shfl_xor(m, o, 32));
  float s = expf(v0 - m) + expf(v1 - m);
#pragma unroll
  for (int o = 16; o; o >>= 1) s += __shfl_xor(s, o, 32);
  float lse = m + logf(s);
  row[lane] = v0 - lse;
  row[lane + 32] = v1 - lse;
}

// ---------------------------------------------------------------------------
// Host-side orchestration
// ---------------------------------------------------------------------------
extern "C" void kernel_launch(void* const* d_in, const int* in_sizes, int n_in,
                              void* d_out, int out_size, void* d_ws, size_t ws_size,
                              hipStream_t stream) {
  const float* x  = (const float*)d_in[0];
  const int*   ei = (const int*)d_in[1];
  const float* W1 = (const float*)d_in[2];
  const float* b1 = (const float*)d_in[3];
  const float* W2 = (const float*)d_in[4];
  const float* b2 = (const float*)d_in[5];
  float* out = (float*)d_out;

  const int F = 128, Hh = 128, Cc = 64;
  const int n = in_sizes[0] / F;       // 100000
  const int E = in_sizes[1] / 2;       // 1600000
  const int* src = ei;                 // edge_index[0]
  const int* dst = ei + E;             // edge_index[1]

  // Workspace layout (everything written before read; ws is poison-filled)
  char* ws = (char*)d_ws;
  size_t off = 0;
  float* dinv = (float*)(ws + off);
  off += (((size_t)n * sizeof(float)) + 255) & ~(size_t)255;
  float* bufH = (float*)(ws + off);    // h = x@W1, later g = h1@W2 (first n*64)
  off += (((size_t)n * Hh * sizeof(float)) + 255) & ~(size_t)255;
  float* bufA = (float*)(ws + off);    // layer-1 accumulator -> h1

  const int TPB = 256;
  const long long nH = (long long)n * Hh;

  // 1) degrees -> dinv  (needed by the fused GEMM epilogue)
  k_deg_init<<<(n + TPB - 1) / TPB, TPB, 0, stream>>>(dinv, n);
  k_deg_count<<<(E + TPB - 1) / TPB, TPB, 0, stream>>>(dst, dinv, E);
  k_dinv<<<(n + TPB - 1) / TPB, TPB, 0, stream>>>(dinv, n);

  // 2) h = x@W1 -> bufH ; bufA = h * dinv^2 (self-loop init), fused
  //    4 waves/block x 32 cols = 128 = Hh
  k_gemm_wmma<128, 128><<<n / 16, 128, 0, stream>>>(x, W1, dinv, bufH, bufA);

  // 3) scatter-add edges into bufA
  k_edge_agg<128><<<(E + 7) / 8, TPB, 0, stream>>>(src, dst, dinv, bufH, bufA, E);

  // 4) h1 = relu(bufA + b1), in place
  k_bias_relu<<<(int)((nH + TPB - 1) / TPB), TPB, 0, stream>>>(bufA, b1, nH, Hh - 1);

  // 5) g = h1@W2 -> bufH ; d_out = g * dinv^2 (self-loop init), fused
  //    2 waves/block x 32 cols = 64 = Cc
  k_gemm_wmma<128, 64><<<n / 16, 64, 0, stream>>>(bufA, W2, dinv, bufH, out);

  // 6) scatter-add edges into d_out
  k_edge_agg<64><<<(E + 7) / 8, TPB, 0, stream>>>(src, dst, dinv, bufH, out, E);

  // 7) logits = out + b2 ; out = log_softmax(logits), fused, in place
  k_bias_logsoftmax64<<<(n + 7) / 8, TPB, 0, stream>>>(out, b2, n);
}